// GINConv_20469814133017
// MI455X (gfx1250) — compile-verified
//
#include <hip/hip_runtime.h>
#include <hip/hip_bf16.h>

// ---------------------------------------------------------------------------
// GINConv-style layer on MI455X (gfx1250, wave32, WMMA):
//   agg[row] += x[col]  (edge scatter, L2-resident f32 atomics)
//   h  = x + agg
//   h1 = relu(h @ W1 + b1)        -> v_wmma_f32_16x16x4_f32
//   h2 = h1 @ W2 + b2             -> v_wmma_f32_16x16x4_f32 (+ fused BN stats)
//   out = gamma*(h2-mean)*rsqrt(var+eps)+beta  (folded to out = h2*a + b)
//
// GEMM: WG = 256 threads (8 waves); each wave owns one 16-col tile and keeps
// its 64 B-matrix fragment floats in registers for the whole kernel. WG loops
// over T=5 row tiles; per tile the 16x128 A block is staged in LDS (stride
// 132 -> conflict-free ds_load_2addr_b64 column reads). Inner loop is pure
// LDS-load + WMMA, no global traffic.
// ---------------------------------------------------------------------------

#define NODES   100000
#define EDGES   1600000
#define DIM     128
#define ROWTILE 16
#define NTILES  (NODES / ROWTILE)   // 6250
#define MTILES  5                   // row tiles per workgroup
#define NBLOCKS (NTILES / MTILES)   // 1250
#define LDS_STR 132                 // 128 + 4 pad -> conflict-free column reads

typedef __attribute__((ext_vector_type(2))) float v2f;
typedef __attribute__((ext_vector_type(8))) float v8f;

static __device__ __forceinline__ v8f wmma_f32_16x16x4(v2f a, v2f b, v8f c) {
    // D = A(16x4) * B(4x16) + C(16x16), all fp32, wave32
    return __builtin_amdgcn_wmma_f32_16x16x4_f32(
        /*neg_a=*/false, a, /*neg_b=*/false, b,
        /*c_mod=*/(short)0, c, /*reuse_a=*/false, /*reuse_b=*/false);
}

static __device__ __forceinline__ void atomic_add_f32(float* p, float v) {
    unsafeAtomicAdd(p, v);  // global_atomic_add_f32 (no CAS loop)
}

// ---- kernel 1: buf_h = x (residual/self term), zero the stats slots --------
__global__ __launch_bounds__(256) void init_kernel(const float* __restrict__ x,
                                                   float* __restrict__ buf_h,
                                                   float* __restrict__ stats) {
    const int t = blockIdx.x * 256 + threadIdx.x;          // float4 index
    const float4 v = reinterpret_cast<const float4*>(x)[t];
    reinterpret_cast<float4*>(buf_h)[t] = v;
    if (t < 256) stats[t] = 0.0f;                          // colsum + colsumsq
}

// ---- kernel 2: edge scatter-add, one wave per edge -------------------------
__global__ __launch_bounds__(256) void edge_kernel(const float* __restrict__ x,
                                                   const long long* __restrict__ ei,
                                                   float* __restrict__ buf_h) {
    const int t    = blockIdx.x * 256 + threadIdx.x;
    const int e    = t >> 5;                               // edge id
    const int lane = t & 31;                               // 4 floats per lane
    const int row  = (int)ei[e];                           // edge_index[0][e]
    const int col  = (int)ei[EDGES + e];                   // edge_index[1][e]
    const float4 v = *reinterpret_cast<const float4*>(x + (size_t)col * DIM + lane * 4);
    float* dst     = buf_h + (size_t)row * DIM + lane * 4;
    atomic_add_f32(dst + 0, v.x);
    atomic_add_f32(dst + 1, v.y);
    atomic_add_f32(dst + 2, v.z);
    atomic_add_f32(dst + 3, v.w);
}

// ---- kernels 3/4: multi-row-tile GEMM, register-resident B fragments -------
template <bool RELU, bool STATS>
__global__ __launch_bounds__(256) void gemm_kernel(const float* __restrict__ H,
                                                   const float* __restrict__ W,
                                                   const float* __restrict__ bias,
                                                   float* __restrict__ OUT,
                                                   float* __restrict__ colsum,
                                                   float* __restrict__ colsumsq) {
    __shared__ float Atile[ROWTILE * LDS_STR];
    __shared__ float lsum[DIM];
    __shared__ float lsq[DIM];

    const int tid  = threadIdx.x;
    const int wave = tid >> 5;
    const int lane = tid & 31;
    const int half = lane >> 4;        // K-half / M-half selector
    const int idx  = lane & 15;
    const int col0 = wave * 16;        // this wave's 16-column tile

    if (STATS && tid < DIM) { lsum[tid] = 0.0f; lsq[tid] = 0.0f; }

    // Preload this wave's B fragments for all 32 K-steps (64 VGPRs/lane).
    v2f bfrag[DIM / 4];
#pragma unroll
    for (int k = 0; k < DIM / 4; ++k) {
        const int ka = 4 * k + 2 * half;                   // B[ka..ka+1][col]
        bfrag[k].x = W[(size_t)ka * DIM + col0 + idx];
        bfrag[k].y = W[(size_t)(ka + 1) * DIM + col0 + idx];
    }
    const float bv = bias[col0 + idx];

    float s = 0.0f, q = 0.0f;          // BN stats for column col0+idx

#pragma unroll 1
    for (int mt = 0; mt < MTILES; ++mt) {
        const int row0 = (blockIdx.x * MTILES + mt) * ROWTILE;

        __syncthreads();               // protect previous tile's LDS reads
        // cooperative load of the 16x128 A tile (512 float4, 2 per thread)
#pragma unroll
        for (int i = 0; i < 2; ++i) {
            const int f  = tid + i * 256;                  // float4 index
            const int r  = f >> 5;
            const int cg = f & 31;
            const float4 v =
                *reinterpret_cast<const float4*>(H + (size_t)(row0 + r) * DIM + cg * 4);
            float* dst = &Atile[r * LDS_STR + cg * 4];
            dst[0] = v.x; dst[1] = v.y; dst[2] = v.z; dst[3] = v.w;
        }
        __syncthreads();

        v8f acc = {};
#pragma unroll
        for (int k = 0; k < DIM / 4; ++k) {
            const int ka = 4 * k + 2 * half;               // this half's K pair
            v2f a;                                         // A[M=idx][ka..ka+1]
            a.x = Atile[idx * LDS_STR + ka];
            a.y = Atile[idx * LDS_STR + ka + 1];
            acc = wmma_f32_16x16x4(a, bfrag[k], acc);
        }

#pragma unroll
        for (int v = 0; v < 8; ++v) {
            float r = acc[v] + bv;     // lane holds (M = v + 8*half, N = idx)
            if (RELU) r = r > 0.0f ? r : 0.0f;
            OUT[(size_t)(row0 + v + 8 * half) * DIM + col0 + idx] = r;
            if (STATS) { s += r; q += r * r; }
        }
    }

    if (STATS) {
        atomicAdd(&lsum[col0 + idx], s);   // ds_add_f32, 2 lanes/address
        atomicAdd(&lsq[col0 + idx], q);
        __syncthreads();
        if (tid < DIM) {
            atomic_add_f32(&colsum[tid], lsum[tid]);
            atomic_add_f32(&colsumsq[tid], lsq[tid]);
        }
    }
}

// ---- kernel 5: fold BN into out = h2*a + b ---------------------------------
__global__ __launch_bounds__(128) void finalize_kernel(const float* __restrict__ colsum,
                                                       const float* __restrict__ colsumsq,
                                                       const float* __restrict__ gamma,
                                                       const float* __restrict__ beta,
                                                       float* __restrict__ ab) {
    const int c = threadIdx.x;
    const float invn = 1.0f / (float)NODES;
    const float mean = colsum[c] * invn;
    const float var  = colsumsq[c] * invn - mean * mean;   // biased, like torch BN
    const float a    = gamma[c] * rsqrtf(var + 1e-5f);
    ab[c]       = a;
    ab[DIM + c] = beta[c] - mean * a;
}

// ---- kernel 6: in-place BN epilogue on d_out -------------------------------
__global__ __launch_bounds__(256) void norm_kernel(float* __restrict__ out,
                                                   const float* __restrict__ ab) {
    const int t = blockIdx.x * 256 + threadIdx.x;          // float4 index
    const int c = (t & 31) * 4;                            // column base
    float4 v = reinterpret_cast<float4*>(out)[t];
    v.x = v.x * ab[c + 0] + ab[DIM + c + 0];
    v.y = v.y * ab[c + 1] + ab[DIM + c + 1];
    v.z = v.z * ab[c + 2] + ab[DIM + c + 2];
    v.w = v.w * ab[c + 3] + ab[DIM + c + 3];
    reinterpret_cast<float4*>(out)[t] = v;
}

extern "C" void kernel_launch(void* const* d_in, const int* in_sizes, int n_in,
                              void* d_out, int out_size, void* d_ws, size_t ws_size,
                              hipStream_t stream) {
    const float*     x     = (const float*)d_in[0];
    const long long* ei    = (const long long*)d_in[1];   // int64 (2,E)
    const float*     W1    = (const float*)d_in[2];
    const float*     b1    = (const float*)d_in[3];
    const float*     W2    = (const float*)d_in[4];
    const float*     b2    = (const float*)d_in[5];
    const float*     gamma = (const float*)d_in[6];
    const float*     beta  = (const float*)d_in[7];
    float* out = (float*)d_out;

    float* buf_h    = (float*)d_ws;                        // N*128
    float* buf_h1   = buf_h + (size_t)NODES * DIM;         // N*128
    float* colsum   = buf_h1 + (size_t)NODES * DIM;        // 128
    float* colsumsq = colsum + DIM;                        // 128
    float* ab       = colsum + 2 * DIM;                    // 256 (a, b)

    const int elem4_blocks = (NODES * DIM / 4) / 256;      // 12500

    init_kernel<<<elem4_blocks, 256, 0, stream>>>(x, buf_h, colsum);
    edge_kernel<<<(EDGES * 32) / 256, 256, 0, stream>>>(x, ei, buf_h);
    gemm_kernel<true,  false><<<NBLOCKS, 256, 0, stream>>>(buf_h,  W1, b1, buf_h1, nullptr, nullptr);
    gemm_kernel<false, true ><<<NBLOCKS, 256, 0, stream>>>(buf_h1, W2, b2, out,    colsum,  colsumsq);
    finalize_kernel<<<1, 128, 0, stream>>>(colsum, colsumsq, gamma, beta, ab);
    norm_kernel<<<elem4_blocks, 256, 0, stream>>>(out, ab);
}